// KBGATModel_61057255080433
// MI455X (gfx1250) — compile-verified
//
#include <hip/hip_runtime.h>
#include <math.h>

// ---------------- problem constants (match reference) ----------------
static constexpr int D50  = 50;
static constexpr int NENT = 100000;
static constexpr int NREL = 500;
static constexpr int WPB  = 4;   // waves (16-edge tiles) per block in the WMMA pass

typedef __attribute__((ext_vector_type(2))) float v2f;
typedef __attribute__((ext_vector_type(8))) float v8f;

// ---------------- order-preserving float<->uint map for atomic max ----------------
__device__ __forceinline__ unsigned mapOrd(float x) {
  unsigned b = __float_as_uint(x);
  return (b & 0x80000000u) ? ~b : (b | 0x80000000u);
}
__device__ __forceinline__ float unmapOrd(unsigned u) {
  unsigned b = (u & 0x80000000u) ? (u & 0x7fffffffu) : ~u;
  return __uint_as_float(b);
}
// mapOrd(-inf) == 0x007fffff
#define MAPPED_NEG_INF 0x007fffffu

// ---------------- init: zero accumulator, reset segment max/sum ----------------
__global__ void k_init(float* __restrict__ acc, long nAcc,
                       unsigned* __restrict__ mA, float* __restrict__ sA, int nSeg) {
  long i = (long)blockIdx.x * blockDim.x + threadIdx.x;
  if (i < nAcc) acc[i] = 0.f;
  if (i < nSeg) { mA[i] = MAPPED_NEG_INF; sA[i] = 0.f; }
}

// ---------------- wa = W @ a  (150-vector) ----------------
__global__ void k_wa(const float* __restrict__ W, const float* __restrict__ avec,
                     float* __restrict__ wa) {
  int i = blockIdx.x * blockDim.x + threadIdx.x;
  if (i < 3 * D50) {
    float s = 0.f;
    for (int j = 0; j < D50; ++j) s = fmaf(W[i * D50 + j], avec[j], s);
    wa[i] = s;
  }
}

// ---------------- rel = (relu?)(Rin @ Wr)  [500 x 50] ----------------
__global__ void k_rel(const float* __restrict__ Rin, const float* __restrict__ Wr,
                      float* __restrict__ out, int n, int doRelu) {
  int i = blockIdx.x * blockDim.x + threadIdx.x;
  if (i >= n) return;
  int r = i / D50, j = i - r * D50;
  float s = 0.f;
  for (int k = 0; k < D50; ++k) s = fmaf(Rin[r * D50 + k], Wr[k * D50 + j], s);
  out[i] = (doRelu && s < 0.f) ? 0.f : s;
}

// ---------------- pass 1: score = leaky(concat . wa), atomic segment max ----------------
__global__ void k_score_max(const int* __restrict__ hI, const int* __restrict__ rI,
                            const int* __restrict__ tI,
                            const float* __restrict__ Ef, const float* __restrict__ Rf,
                            const float* __restrict__ wa,
                            unsigned* __restrict__ mA, int nE) {
  int e = blockIdx.x * blockDim.x + threadIdx.x;
  if (e >= nE) return;
  const float* xh = Ef + (long)hI[e] * D50;
  const float* xr = Rf + (long)rI[e] * D50;
  const float* xt = Ef + (long)tI[e] * D50;
  float s = 0.f;
  #pragma unroll 10
  for (int j = 0; j < D50; ++j) {
    s = fmaf(xh[j], wa[j], s);
    s = fmaf(xr[j], wa[D50 + j], s);
    s = fmaf(xt[j], wa[100 + j], s);
  }
  float sc = s > 0.f ? s : 0.2f * s;
  atomicMax(&mA[hI[e]], mapOrd(sc));
}

// ---------------- pass 2: WMMA feat tiles, exp-weighted scatter ----------------
// Each wave owns one 16-edge tile: feat[16x50] = A[16x150] @ W[150x50],
// computed as 3 K-segments of 50 (padded to 52) with V_WMMA_F32_16X16X4_F32.
// Waves are fully decoupled: per-wave LDS regions + same-wave in-order DS ops,
// so only compiler scheduling fences (wave_barrier) are needed inside the loop.
__global__ __launch_bounds__(WPB * 32) void k_attn(
    const int* __restrict__ hI, const int* __restrict__ rI, const int* __restrict__ tI,
    const float* __restrict__ Ef, const float* __restrict__ Rf,
    const float* __restrict__ W, const float* __restrict__ avec,
    const unsigned* __restrict__ mA, float* __restrict__ sA,
    float* __restrict__ acc, int nE) {
  // W transposed: Wt[n][k], n in [0,64), k in [0,152), zero-padded (read-only after init)
  __shared__ __align__(16) float Wt[64 * 152];
  // per-wave A tile (later reused as the feat tile): [16 rows][64 cols]
  __shared__ __align__(16) float Av[WPB][16 * 64];
  __shared__ float aV[64];
  __shared__ int   idxs[WPB][48];
  __shared__ float eV[WPB][16];

  const int tid = threadIdx.x;
  for (int i = tid; i < 64 * 152; i += WPB * 32) {
    int n = i / 152, k = i - n * 152;
    Wt[i] = (n < D50 && k < 3 * D50) ? W[k * D50 + n] : 0.f;
  }
  if (tid < 64) aV[tid] = (tid < D50) ? avec[tid] : 0.f;
  __syncthreads();   // Wt/aV visible to all waves; the only block-wide barrier

  const int w    = tid >> 5;
  const int lane = tid & 31;
  const int m16  = lane & 15;   // row (A/M) or column (B/N) within fragment
  const int hi   = lane >> 4;   // K sub-pair selector: kbase = 2*hi
  const int nTiles = (nE + 15) >> 4;

  for (int tile = blockIdx.x * WPB + w; tile < nTiles; tile += gridDim.x * WPB) {
    const int base = tile << 4;

    if (lane < 16) {
      int ei = base + lane;
      bool v = ei < nE;
      idxs[w][lane]      = v ? hI[ei] : 0;
      idxs[w][16 + lane] = v ? rI[ei] : 0;
      idxs[w][32 + lane] = v ? tI[ei] : 0;
    }
    __builtin_amdgcn_wave_barrier();   // same-wave DS in-order; fence compiler only

    v8f acc0 = {}, acc1 = {}, acc2 = {}, acc3 = {};
    for (int seg = 0; seg < 3; ++seg) {
      const float* src = (seg == 1) ? Rf : Ef;
      const int ioff = seg * 16;
      for (int ii = lane; ii < 16 * 52; ii += 32) {
        int row = ii / 52, col = ii - row * 52;
        Av[w][row * 64 + col] =
            (col < D50) ? src[(long)idxs[w][ioff + row] * D50 + col] : 0.f;
      }
      __builtin_amdgcn_wave_barrier();

      const int ks = seg * D50;
      #pragma unroll
      for (int k0 = 0; k0 < 52; k0 += 4) {
        // A fragment: lane holds A[m16][k0 + 2*hi + {0,1}]
        v2f af = *(const v2f*)&Av[w][m16 * 64 + k0 + 2 * hi];
        // B fragments: lane holds B[k0+2*hi+{0,1}][nt*16 + m16] = Wt[n][k]
        v2f b0 = *(const v2f*)&Wt[(0 * 16 + m16) * 152 + ks + k0 + 2 * hi];
        v2f b1 = *(const v2f*)&Wt[(1 * 16 + m16) * 152 + ks + k0 + 2 * hi];
        v2f b2 = *(const v2f*)&Wt[(2 * 16 + m16) * 152 + ks + k0 + 2 * hi];
        v2f b3 = *(const v2f*)&Wt[(3 * 16 + m16) * 152 + ks + k0 + 2 * hi];
        acc0 = __builtin_amdgcn_wmma_f32_16x16x4_f32(false, af, false, b0, (short)0, acc0, false, false);
        acc1 = __builtin_amdgcn_wmma_f32_16x16x4_f32(false, af, false, b1, (short)0, acc1, false, false);
        acc2 = __builtin_amdgcn_wmma_f32_16x16x4_f32(false, af, false, b2, (short)0, acc2, false, false);
        acc3 = __builtin_amdgcn_wmma_f32_16x16x4_f32(false, af, false, b3, (short)0, acc3, false, false);
      }
      __builtin_amdgcn_wave_barrier();   // fragment loads ordered before next staging
    }

    // write feat tile back to LDS (C layout: VGPR v -> M = v + 8*hi, N = m16 + 16*nt)
    #pragma unroll
    for (int v = 0; v < 8; ++v) {
      const int rowm = v + 8 * hi;
      Av[w][rowm * 64 + 0 * 16 + m16] = acc0[v];
      Av[w][rowm * 64 + 1 * 16 + m16] = acc1[v];
      Av[w][rowm * 64 + 2 * 16 + m16] = acc2[v];
      Av[w][rowm * 64 + 3 * 16 + m16] = acc3[v];
    }
    __builtin_amdgcn_wave_barrier();

    if (lane < 16) {
      int ei = base + lane;
      if (ei < nE) {
        float sc = 0.f;
        for (int j = 0; j < D50; ++j) sc = fmaf(Av[w][lane * 64 + j], aV[j], sc);
        sc = sc > 0.f ? sc : 0.2f * sc;
        int hh = idxs[w][lane];
        float fm = unmapOrd(mA[hh]);
        if (!(fm > -3.0e38f)) fm = 0.f;   // mirror reference isfinite guard
        float ee = expf(sc - fm);
        eV[w][lane] = ee;
        atomicAdd(&sA[hh], ee);
      } else {
        eV[w][lane] = 0.f;
      }
    }
    __builtin_amdgcn_wave_barrier();

    // scatter e * feat into per-head accumulator (normalized later)
    for (int row = 0; row < 16; ++row) {
      int ei = base + row;
      if (ei >= nE) break;
      float ee = eV[w][row];
      int hh = idxs[w][row];
      atomicAdd(&acc[(long)hh * D50 + lane], ee * Av[w][row * 64 + lane]);
      if (lane < D50 - 32)
        atomicAdd(&acc[(long)hh * D50 + 32 + lane], ee * Av[w][row * 64 + 32 + lane]);
    }
    __builtin_amdgcn_wave_barrier();   // scatter reads done before next tile restages Av
  }
}

// ---------------- normalize: acc /= (s + 1e-9) ----------------
__global__ void k_norm(float* __restrict__ acc, const float* __restrict__ sA, long n) {
  long i = (long)blockIdx.x * blockDim.x + threadIdx.x;
  if (i < n) acc[i] = acc[i] / (sA[i / D50] + 1e-9f);
}

// ---------------- final: out = acc/(s+1e-9) + E @ Wd + bd ----------------
__global__ void k_final(float* __restrict__ out, const float* __restrict__ sA,
                        const float* __restrict__ E, const float* __restrict__ Wd,
                        const float* __restrict__ bd, long n) {
  long i = (long)blockIdx.x * blockDim.x + threadIdx.x;
  if (i >= n) return;
  long ent = i / D50;
  int j = (int)(i - ent * D50);
  float dot = 0.f;
  for (int k = 0; k < D50; ++k) dot = fmaf(E[ent * D50 + k], Wd[k * D50 + j], dot);
  out[i] = out[i] / (sA[ent] + 1e-9f) + dot + bd[j];
}

// ---------------- launcher ----------------
extern "C" void kernel_launch(void* const* d_in, const int* in_sizes, int n_in,
                              void* d_out, int out_size, void* d_ws, size_t ws_size,
                              hipStream_t stream) {
  const int*   hI  = (const int*)d_in[0];
  const int*   rI  = (const int*)d_in[1];
  const int*   tI  = (const int*)d_in[2];
  const float* E   = (const float*)d_in[3];
  const float* R   = (const float*)d_in[4];
  const float* W0  = (const float*)d_in[5];
  const float* a0  = (const float*)d_in[6];
  const float* Wr0 = (const float*)d_in[7];
  const float* W1  = (const float*)d_in[8];
  const float* a1  = (const float*)d_in[9];
  const float* Wr1 = (const float*)d_in[10];
  const float* Wd  = (const float*)d_in[11];
  const float* bd  = (const float*)d_in[12];
  const int nE = in_sizes[0];

  float* outEnt = (float*)d_out;                       // [NENT*50]
  float* outRel = (float*)d_out + (long)NENT * D50;    // [NREL*50]

  // workspace layout (bytes): e0 | rel0 | wa0 | wa1 | m | s  (~20.9 MB)
  char* ws = (char*)d_ws;
  float*    e0   = (float*)ws;                       // 5,000,000 f32
  float*    rel0 = (float*)(ws + 20000000);          //    25,000 f32
  float*    wa0  = (float*)(ws + 20100096);          //       160 f32
  float*    wa1  = (float*)(ws + 20100736);          //       160 f32
  unsigned* mA   = (unsigned*)(ws + 20101376);       //   100,000 u32
  float*    sAr  = (float*)(ws + 20501376);          //   100,000 f32

  const long nAcc  = (long)NENT * D50;
  const int gInit  = (int)((nAcc + 255) / 256);
  const int gEdge  = (nE + 255) / 256;
  const int nTiles = (nE + 15) / 16;
  const int gAttn  = (nTiles + WPB - 1) / WPB;
  const int gRel   = (NREL * D50 + 127) / 128;

  // ---- layer 0 ----
  k_wa<<<2, 128, 0, stream>>>(W0, a0, wa0);
  k_rel<<<gRel, 128, 0, stream>>>(R, Wr0, rel0, NREL * D50, 1);
  k_init<<<gInit, 256, 0, stream>>>(e0, nAcc, mA, sAr, NENT);
  k_score_max<<<gEdge, 256, 0, stream>>>(hI, rI, tI, E, R, wa0, mA, nE);
  k_attn<<<gAttn, WPB * 32, 0, stream>>>(hI, rI, tI, E, R, W0, a0, mA, sAr, e0, nE);
  k_norm<<<gInit, 256, 0, stream>>>(e0, sAr, nAcc);

  // ---- layer 1 ----
  k_wa<<<2, 128, 0, stream>>>(W1, a1, wa1);
  k_rel<<<gRel, 128, 0, stream>>>(rel0, Wr1, outRel, NREL * D50, 0);
  k_init<<<gInit, 256, 0, stream>>>(outEnt, nAcc, mA, sAr, NENT);
  k_score_max<<<gEdge, 256, 0, stream>>>(hI, rI, tI, e0, rel0, wa1, mA, nE);
  k_attn<<<gAttn, WPB * 32, 0, stream>>>(hI, rI, tI, e0, rel0, W1, a1, mA, sAr, outEnt, nE);
  k_final<<<gInit, 256, 0, stream>>>(outEnt, sAr, E, Wd, bd, nAcc);
}